// OMP_7550552506949
// MI455X (gfx1250) — compile-verified
//
#include <hip/hip_runtime.h>
#include <hip/hip_bf16.h>

// ---------------------------------------------------------------------------
// Problem constants (from reference)
// ---------------------------------------------------------------------------
#define BATCH   2048
#define NSIG    64          // SIGNAL_DIM
#define FRQ     4096        // FR_SIZE
#define LWIN    20          // PARAM
#define MWIN    45          // NSIG - LWIN + 1
#define KMAX    16
#define NFB     8           // frequency blocks for corr/argmax kernel
#define FBLK    (FRQ / NFB) // 512 freqs per block

typedef float v2f __attribute__((ext_vector_type(2)));
typedef float v8f __attribute__((ext_vector_type(8)));

// CDNA5 async global->LDS path (ASYNCcnt-tracked DMA-style copy), guarded so
// the kernel still builds with the synchronous fallback on older toolchains.
#if __has_builtin(__builtin_amdgcn_global_load_async_to_lds_b64) && \
    __has_builtin(__builtin_amdgcn_s_wait_asynccnt)
#define USE_ASYNC_LDS 1
// Exact parameter types from the clang diagnostic:
//   param 1: __attribute__((__vector_size__(2*sizeof(int)))) int __device__ *
typedef int async_b64_t __attribute__((__vector_size__(2 * sizeof(int))));
typedef __attribute__((address_space(1))) async_b64_t* async_gptr;
typedef __attribute__((address_space(3))) async_b64_t* async_lptr;
#else
#define USE_ASYNC_LDS 0
#endif

// ---------------------------------------------------------------------------
// Kernel 0: zero output + global-max scalar
// ---------------------------------------------------------------------------
__global__ void k_zero(float* __restrict__ out, unsigned* __restrict__ gmax) {
    size_t i = (size_t)blockIdx.x * blockDim.x + threadIdx.x;
    if (i == 0) *gmax = 0u;
    const size_t n = (size_t)BATCH * FRQ;
    const size_t stride = (size_t)gridDim.x * blockDim.x;
    for (; i < n; i += stride) out[i] = 0.0f;
}

// ---------------------------------------------------------------------------
// Kernel 1: SORTE model-order estimation + residual init (1 wave / batch)
// 20x20 Hermitian R embedded as real-symmetric 40x40, cyclic Jacobi eigvals.
// ---------------------------------------------------------------------------
__global__ void __launch_bounds__(32) k_sorte(const float* __restrict__ x,
                                              float* __restrict__ r_re,
                                              float* __restrict__ r_im,
                                              int* __restrict__ nf) {
    __shared__ float xr[NSIG], xi[NSIG];
    __shared__ float S[40][41];
    __shared__ float srt[40];
    const int lane = threadIdx.x;
    const int b = blockIdx.x;
    const float* xb = x + (size_t)b * 2 * NSIG;

    for (int k = lane; k < NSIG; k += 32) {
        float vr = xb[k], vi = xb[NSIG + k];
        xr[k] = vr; xi[k] = vi;
        r_re[(size_t)b * NSIG + k] = vr;   // residual starts as xc
        r_im[(size_t)b * NSIG + k] = vi;
    }
    __syncthreads();

    // R[l][k] = sum_m x[l+m] * conj(x[k+m]) / M ; real-symmetric embedding
    for (int e = lane; e < LWIN * LWIN; e += 32) {
        int l = e / LWIN, k = e % LWIN;
        float re = 0.f, im = 0.f;
        for (int m = 0; m < MWIN; ++m) {
            float ar = xr[l + m], ai = xi[l + m];
            float br = xr[k + m], bi = xi[k + m];
            re += ar * br + ai * bi;     // x * conj(y)
            im += ai * br - ar * bi;
        }
        re *= (1.0f / MWIN); im *= (1.0f / MWIN);
        S[l][k] = re;               S[l + LWIN][k + LWIN] = re;
        S[l][k + LWIN] = -im;       S[l + LWIN][k] = im;
    }
    __syncthreads();

    // Cyclic Jacobi (rotation params computed redundantly on all lanes)
    for (int sweep = 0; sweep < 8; ++sweep) {
        for (int p = 0; p < 39; ++p) {
            for (int q = p + 1; q < 40; ++q) {
                float apq = S[p][q];
                if (fabsf(apq) > 1e-12f) {
                    float app = S[p][p], aqq = S[q][q];
                    float tau = (aqq - app) / (2.0f * apq);
                    float tt = (tau >= 0.f ? 1.f : -1.f) /
                               (fabsf(tau) + sqrtf(1.f + tau * tau));
                    float c = rsqrtf(1.f + tt * tt);
                    float s = tt * c;
                    for (int j = lane; j < 40; j += 32) {
                        float sp = S[p][j], sq = S[q][j];
                        S[p][j] = c * sp - s * sq;
                        S[q][j] = s * sp + c * sq;
                    }
                    __syncthreads();
                    for (int j = lane; j < 40; j += 32) {
                        float sp = S[j][p], sq = S[j][q];
                        S[j][p] = c * sp - s * sq;
                        S[j][q] = s * sp + c * sq;
                    }
                    __syncthreads();
                }
            }
        }
    }

    // rank-sort the 40 diagonal values descending (eigvals appear twice)
    for (int e = lane; e < 40; e += 32) {
        float v = S[e][e];
        int rank = 0;
        for (int j = 0; j < 40; ++j) {
            float vj = S[j][j];
            rank += (vj > v) || (vj == v && j < e);
        }
        srt[rank] = v;
    }
    __syncthreads();

    if (lane == 0) {
        float lam[LWIN];
        for (int i = 0; i < LWIN; ++i) lam[i] = srt[2 * i];
        float d[LWIN - 1];
        for (int i = 0; i < LWIN - 1; ++i) d[i] = lam[i] - lam[i + 1];
        float var[LWIN - 1];
        float s1 = 0.f, s2 = 0.f;
        for (int i = LWIN - 2; i >= 0; --i) {       // suffix means/vars
            s1 += d[i]; s2 += d[i] * d[i];
            float cnt = (float)(LWIN - 1 - i);
            float mean = s1 / cnt;
            float v = s2 / cnt - mean * mean;
            var[i] = v > 0.f ? v : 0.f;
        }
        const float tiny = 1e-12f;
        float bestr = __builtin_inff(); int besti = 0;
        for (int i = 0; i < LWIN - 2; ++i) {
            float r = (var[i] > tiny) ? (var[i + 1] / fmaxf(var[i], tiny))
                                      : __builtin_inff();
            if (r < bestr) { bestr = r; besti = i; }
        }
        int nfv = besti + 1;
        nf[b] = nfv < 1 ? 1 : (nfv > KMAX ? KMAX : nfv);
    }
}

// ---------------------------------------------------------------------------
// Kernel 2: fused complex GEMM corr = r @ conj(D) + per-row argmax, via
// v_wmma_f32_16x16x4_f32.  1 wave per (16-batch tile, 512-freq block).
//   corr_re = rr*Dr + ri*Di ; corr_im = ri*Dr - rr*Di   (conj(D))
// ---------------------------------------------------------------------------
__global__ void __launch_bounds__(32) k_corr(const float* __restrict__ Dre,
                                             const float* __restrict__ Dim,
                                             const float* __restrict__ r_re,
                                             const float* __restrict__ r_im,
                                             float* __restrict__ bval,
                                             int* __restrict__ bidx) {
    __shared__ v2f Ar[16][33];   // padded: bank-spread for lane=row access
    __shared__ v2f Ai[16][33];
    const int lane = threadIdx.x;
    const int b0 = blockIdx.x * 16;          // batch tile
    const int fb = blockIdx.y;               // frequency block

    // Stage A block (16 rows x 64 K) as K-pairs.
#if USE_ASYNC_LDS
    // CDNA5 path: direct global->LDS async copies (ASYNCcnt), no VGPR bounce.
    for (int e = lane; e < 16 * 32; e += 32) {
        int m = e >> 5, p = e & 31;
        const float* gr_ = r_re + (size_t)(b0 + m) * NSIG + 2 * p;
        const float* gi_ = r_im + (size_t)(b0 + m) * NSIG + 2 * p;
        __builtin_amdgcn_global_load_async_to_lds_b64(
            (async_gptr)gr_, (async_lptr)&Ar[m][p], 0, 0);
        __builtin_amdgcn_global_load_async_to_lds_b64(
            (async_gptr)gi_, (async_lptr)&Ai[m][p], 0, 0);
    }
    __builtin_amdgcn_s_wait_asynccnt(0);
    __syncthreads();
#else
    for (int e = lane; e < 16 * 32; e += 32) {
        int m = e >> 5, p = e & 31;
        const float* pr = r_re + (size_t)(b0 + m) * NSIG + 2 * p;
        const float* pi = r_im + (size_t)(b0 + m) * NSIG + 2 * p;
        Ar[m][p] = (v2f){pr[0], pr[1]};
        Ai[m][p] = (v2f){pi[0], pi[1]};
    }
    __syncthreads();
#endif

    const int hi = lane >> 4;    // half-wave selects K pair (K+0/1 vs K+2/3)
    const int lm = lane & 15;    // A: row M ; B/C: column N
    float best[8];  int bix[8];
#pragma unroll
    for (int v = 0; v < 8; ++v) { best[v] = -1.0f; bix[v] = 0; }

    const int n_base = fb * FBLK;
    for (int tile = 0; tile < FBLK / 16; ++tile) {
        const int n0 = n_base + tile * 16;
        const int n = n0 + lm;
        if (n0 + 16 < FRQ) {                         // warm next tile (L2)
            __builtin_prefetch(&Dre[(size_t)(2 * hi) * FRQ + n + 16], 0, 0);
            __builtin_prefetch(&Dim[(size_t)(2 * hi) * FRQ + n + 16], 0, 0);
        }
        v8f accre = {}; v8f accim = {};
#pragma unroll 4
        for (int kc = 0; kc < 16; ++kc) {            // K = 64 in chunks of 4
            v2f are = Ar[lm][2 * kc + hi];
            v2f aim = Ai[lm][2 * kc + hi];
            v2f arn = (v2f){-are.x, -are.y};         // for the -rr*Di term
            const int kk = 4 * kc + 2 * hi;
            v2f bre = (v2f){Dre[(size_t)kk * FRQ + n],
                            Dre[(size_t)(kk + 1) * FRQ + n]};
            v2f bim = (v2f){Dim[(size_t)kk * FRQ + n],
                            Dim[(size_t)(kk + 1) * FRQ + n]};
            accre = __builtin_amdgcn_wmma_f32_16x16x4_f32(
                        false, are, false, bre, (short)0, accre, false, false);
            accre = __builtin_amdgcn_wmma_f32_16x16x4_f32(
                        false, aim, false, bim, (short)0, accre, false, false);
            accim = __builtin_amdgcn_wmma_f32_16x16x4_f32(
                        false, aim, false, bre, (short)0, accim, false, false);
            accim = __builtin_amdgcn_wmma_f32_16x16x4_f32(
                        false, arn, false, bim, (short)0, accim, false, false);
        }
#pragma unroll
        for (int v = 0; v < 8; ++v) {                // |corr|^2, running max
            float m2 = accre[v] * accre[v] + accim[v] * accim[v];
            if (m2 > best[v]) { best[v] = m2; bix[v] = n; }
        }
    }

    // Reduce across the 16 lanes holding each row (tie -> lower index)
#pragma unroll
    for (int off = 8; off >= 1; off >>= 1) {
#pragma unroll
        for (int v = 0; v < 8; ++v) {
            float ov = __shfl_xor(best[v], off, 32);
            int   oi = __shfl_xor(bix[v], off, 32);
            if (ov > best[v] || (ov == best[v] && oi < bix[v])) {
                best[v] = ov; bix[v] = oi;
            }
        }
    }
    if (lm == 0) {   // lane 0 -> rows 0..7 ; lane 16 -> rows 8..15
#pragma unroll
        for (int v = 0; v < 8; ++v) {
            int row = b0 + hi * 8 + v;
            bval[(size_t)row * NFB + fb] = best[v];
            bidx[(size_t)row * NFB + fb] = bix[v];
        }
    }
}

// ---------------------------------------------------------------------------
// Kernel 3: per-batch OMP step: argmax combine, Gram + complex Cholesky
// solve, residual update / final output scatter.  1 wave per batch.
// ---------------------------------------------------------------------------
__global__ void __launch_bounds__(32) k_solve(const float* __restrict__ x,
                                              const float* __restrict__ Dre,
                                              const float* __restrict__ Dim,
                                              const float* __restrict__ bval,
                                              const int* __restrict__ bidx,
                                              const int* __restrict__ nf,
                                              int* __restrict__ sup,
                                              float* __restrict__ r_re,
                                              float* __restrict__ r_im,
                                              float* __restrict__ out,
                                              unsigned* __restrict__ gmax,
                                              int t) {
    const int lane = threadIdx.x;
    const int b = blockIdx.x;
    const int nfb = nf[b];
    if (t >= nfb) return;                 // batch finished (uniform per block)

    __shared__ float sAr[KMAX][NSIG], sAi[KMAX][NSIG];
    __shared__ float Gr[KMAX][KMAX + 1], Gi[KMAX][KMAX + 1];
    __shared__ float Lr[KMAX][KMAX + 1], Li[KMAX][KMAX + 1];
    __shared__ float xr[NSIG], xi[NSIG];
    __shared__ float br_[KMAX], bi_[KMAX], cr_[KMAX], ci_[KMAX];
    __shared__ int s_sup[KMAX];

    // combine block-wise argmax (redundant on all lanes; deterministic)
    {
        float bv = -1.f; int bx = 0;
        for (int f = 0; f < NFB; ++f) {
            float v = bval[(size_t)b * NFB + f];
            int   i = bidx[(size_t)b * NFB + f];
            if (v > bv || (v == bv && i < bx)) { bv = v; bx = i; }
        }
        if (lane == 0) { sup[(size_t)b * KMAX + t] = bx; s_sup[t] = bx; }
    }
    if (lane < t) s_sup[lane] = sup[(size_t)b * KMAX + lane];
    for (int k = lane; k < NSIG; k += 32) {
        xr[k] = x[(size_t)b * 2 * NSIG + k];
        xi[k] = x[(size_t)b * 2 * NSIG + NSIG + k];
    }
    __syncthreads();

    const int ts = t + 1;
    // stage A columns (D at support indices)
    for (int e = lane; e < ts * NSIG; e += 32) {
        int j = e >> 6, k = e & 63;
        int col = s_sup[j];
        sAr[j][k] = Dre[(size_t)k * FRQ + col];
        sAi[j][k] = Dim[(size_t)k * FRQ + col];
    }
    __syncthreads();

    // G = A^H A + eps I ; bvec = A^H xc
    for (int e = lane; e < ts * ts; e += 32) {
        int i = e / ts, j = e % ts;
        float gr = 0.f, gi = 0.f;
        for (int k = 0; k < NSIG; ++k) {
            gr += sAr[i][k] * sAr[j][k] + sAi[i][k] * sAi[j][k];
            gi += sAr[i][k] * sAi[j][k] - sAi[i][k] * sAr[j][k];
        }
        if (i == j) gr += 1e-6f;
        Gr[i][j] = gr; Gi[i][j] = gi;
    }
    if (lane < ts) {
        int j = lane;
        float r0 = 0.f, i0 = 0.f;
        for (int k = 0; k < NSIG; ++k) {
            r0 += sAr[j][k] * xr[k] + sAi[j][k] * xi[k];
            i0 += sAr[j][k] * xi[k] - sAi[j][k] * xr[k];
        }
        br_[j] = r0; bi_[j] = i0;
    }
    __syncthreads();

    // Complex Cholesky G = L L^H, forward/backward solve (tiny; lane 0)
    if (lane == 0) {
        for (int j = 0; j < ts; ++j) {
            float s = Gr[j][j];
            for (int k = 0; k < j; ++k)
                s -= Lr[j][k] * Lr[j][k] + Li[j][k] * Li[j][k];
            float dj = sqrtf(fmaxf(s, 1e-20f));
            Lr[j][j] = dj; Li[j][j] = 0.f;
            float inv = 1.0f / dj;
            for (int i2 = j + 1; i2 < ts; ++i2) {
                float cre = Gr[i2][j], cim = Gi[i2][j];
                for (int k = 0; k < j; ++k) {      // -= L[i][k]*conj(L[j][k])
                    cre -= Lr[i2][k] * Lr[j][k] + Li[i2][k] * Li[j][k];
                    cim -= Li[i2][k] * Lr[j][k] - Lr[i2][k] * Li[j][k];
                }
                Lr[i2][j] = cre * inv; Li[i2][j] = cim * inv;
            }
        }
        for (int i2 = 0; i2 < ts; ++i2) {          // L z = b (in place)
            float zr = br_[i2], zi = bi_[i2];
            for (int k = 0; k < i2; ++k) {
                zr -= Lr[i2][k] * br_[k] - Li[i2][k] * bi_[k];
                zi -= Lr[i2][k] * bi_[k] + Li[i2][k] * br_[k];
            }
            float inv = 1.0f / Lr[i2][i2];
            br_[i2] = zr * inv; bi_[i2] = zi * inv;
        }
        for (int i2 = ts - 1; i2 >= 0; --i2) {     // L^H c = z
            float cre = br_[i2], cim = bi_[i2];
            for (int k = i2 + 1; k < ts; ++k) {    // -= conj(L[k][i])*c[k]
                cre -= Lr[k][i2] * cr_[k] + Li[k][i2] * ci_[k];
                cim -= Lr[k][i2] * ci_[k] - Li[k][i2] * cr_[k];
            }
            float inv = 1.0f / Lr[i2][i2];
            cr_[i2] = cre * inv; ci_[i2] = cim * inv;
        }
    }
    __syncthreads();

    if (t == nfb - 1) {                            // final iteration: emit |c|
        if (lane < ts) {
            float mag = sqrtf(cr_[lane] * cr_[lane] + ci_[lane] * ci_[lane]);
            out[(size_t)b * FRQ + s_sup[lane]] = mag;
            atomicMax(gmax, __float_as_uint(mag)); // mag >= 0: uint order ok
        }
    } else {                                       // residual r = xc - A c
        for (int k = lane; k < NSIG; k += 32) {
            float rr = xr[k], ri = xi[k];
            for (int j = 0; j < ts; ++j) {
                rr -= sAr[j][k] * cr_[j] - sAi[j][k] * ci_[j];
                ri -= sAr[j][k] * ci_[j] + sAi[j][k] * cr_[j];
            }
            r_re[(size_t)b * NSIG + k] = rr;
            r_im[(size_t)b * NSIG + k] = ri;
        }
    }
}

// ---------------------------------------------------------------------------
// Kernel 4: global normalization by max magnitude
// ---------------------------------------------------------------------------
__global__ void k_norm(float* __restrict__ out, const unsigned* __restrict__ gmax) {
    float g = __uint_as_float(*gmax);
    float inv = g > 0.f ? 1.0f / g : 0.f;
    size_t i = (size_t)blockIdx.x * blockDim.x + threadIdx.x;
    if (i < (size_t)BATCH * FRQ) out[i] *= inv;
}

// ---------------------------------------------------------------------------
extern "C" void kernel_launch(void* const* d_in, const int* in_sizes, int n_in,
                              void* d_out, int out_size, void* d_ws, size_t ws_size,
                              hipStream_t stream) {
    (void)in_sizes; (void)n_in; (void)out_size; (void)ws_size;
    const float* x   = (const float*)d_in[0];   // (2048, 2, 64)
    const float* Dre = (const float*)d_in[1];   // (64, 4096)
    const float* Dim = (const float*)d_in[2];   // (64, 4096)
    float* out = (float*)d_out;                 // (2048, 4096)

    // workspace layout (~1.3 MB)
    float* r_re = (float*)d_ws;
    float* r_im = r_re + (size_t)BATCH * NSIG;
    int*   nf   = (int*)(r_im + (size_t)BATCH * NSIG);
    int*   sup  = nf + BATCH;
    float* bval = (float*)(sup + (size_t)BATCH * KMAX);
    int*   bidx = (int*)(bval + (size_t)BATCH * NFB);
    unsigned* gmax = (unsigned*)(bidx + (size_t)BATCH * NFB);

    k_zero<<<4096, 256, 0, stream>>>(out, gmax);
    k_sorte<<<BATCH, 32, 0, stream>>>(x, r_re, r_im, nf);
    for (int t = 0; t < KMAX; ++t) {
        k_corr<<<dim3(BATCH / 16, NFB), 32, 0, stream>>>(Dre, Dim, r_re, r_im,
                                                         bval, bidx);
        k_solve<<<BATCH, 32, 0, stream>>>(x, Dre, Dim, bval, bidx, nf, sup,
                                          r_re, r_im, out, gmax, t);
    }
    k_norm<<<((size_t)BATCH * FRQ) / 256, 256, 0, stream>>>(out, gmax);
}